// EventResidualInjector_62079457296657
// MI455X (gfx1250) — compile-verified
//
#include <hip/hip_runtime.h>
#include <math.h>

#define DD 64

typedef __attribute__((ext_vector_type(2))) float v2f;
typedef __attribute__((ext_vector_type(8))) float v8f;

__device__ __forceinline__ v8f wmma4(v2f a, v2f b, v8f c) {
  // V_WMMA_F32_16X16X4_F32: D = A(16x4) * B(4x16) + C(16x16), fp32
  return __builtin_amdgcn_wmma_f32_16x16x4_f32(false, a, false, b, (short)0, c,
                                               false, false);
}

__device__ __forceinline__ float leaky(float v) { return v > 0.f ? v : 0.2f * v; }
__device__ __forceinline__ float frelu(float v) { return v > 0.f ? v : 0.f; }

// order-preserving float -> uint encoding for atomicMax-based segment max
__device__ __forceinline__ unsigned fenc(float v) {
  unsigned u = __float_as_uint(v);
  return (u & 0x80000000u) ? ~u : (u | 0x80000000u);
}
__device__ __forceinline__ float fdec(unsigned u) {
  u = (u & 0x80000000u) ? (u & 0x7fffffffu) : ~u;
  return __uint_as_float(u);
}

// Wave-level GEMM: X_tile(16 x K, LDS, row-major stride K) * W(K x 64, global row-major)
// accumulated into 4 C/D fragments (16x16 each). A/B fragment layout per CDNA5 ISA
// (16x4 f32 A: lane = m + 16*(k>>1), vgpr = k&1; B symmetric with n).
template <int K>
__device__ __forceinline__ void gemm_tile(const float* Xl,
                                          const float* __restrict__ W,
                                          v8f acc[4]) {
  const int lane = threadIdx.x & 31;
  const int n0 = lane & 15;          // A: row m; B: col n
  const int kb0 = (lane >> 4) << 1;  // 0 or 2
#pragma unroll
  for (int j = 0; j < 4; ++j) {
#pragma unroll 4
    for (int kk = 0; kk < K; kk += 4) {
      const int k0 = kk + kb0;
      v2f a;
      a.x = Xl[n0 * K + k0];
      a.y = Xl[n0 * K + k0 + 1];
      v2f b;
      b.x = W[k0 * DD + j * 16 + n0];
      b.y = W[(k0 + 1) * DD + j * 16 + n0];
      acc[j] = wmma4(a, b, acc[j]);
    }
  }
}

// ---------------------------------------------------------------- init / degree
__global__ void k_init(float* __restrict__ accum, float* __restrict__ deg, int n) {
  long long tid = (long long)blockIdx.x * blockDim.x + threadIdx.x;
  long long tot = (long long)n * DD;
  if (tid < tot) accum[tid] = 0.f;
  if (tid < n) deg[tid] = 1.f;  // self-loop
}

__global__ void k_deg(const int* __restrict__ ei, float* __restrict__ deg, int e) {
  int tid = blockIdx.x * blockDim.x + threadIdx.x;
  if (tid >= e) return;
  atomicAdd(&deg[ei[e + tid]], 1.f);  // dst row of edge_index
}

__global__ void k_dinv(float* __restrict__ deg, int n) {
  int tid = blockIdx.x * blockDim.x + threadIdx.x;
  if (tid < n) deg[tid] = rsqrtf(deg[tid]);  // deg >= 1 always
}

// --------------------------------------------------- encoder MLP + gcn_W (WMMA)
__global__ void __launch_bounds__(128) k_encoder(
    const float* __restrict__ ev, const float* __restrict__ W1,
    const float* __restrict__ b1, const float* __restrict__ W2,
    const float* __restrict__ b2, const float* __restrict__ Wg,
    float* __restrict__ hgcn, int n) {
  __shared__ float sEv[4][16 * 8];
  __shared__ float sHid[4][16 * 64];
  __shared__ float sEmb[4][16 * 64];
  const int wave = threadIdx.x >> 5;
  const int lane = threadIdx.x & 31;
  const int half = lane >> 4;
  const int n0 = lane & 15;
  const int base = (blockIdx.x * 4 + wave) * 16;

  for (int idx = lane; idx < 16 * 8; idx += 32) {
    int m = idx >> 3, k = idx & 7;
    int node = base + m;
    if (node >= n) node = n - 1;
    sEv[wave][idx] = ev[node * 8 + k];
  }
  __syncthreads();

  v8f a1[4] = {};
  gemm_tile<8>(sEv[wave], W1, a1);
#pragma unroll
  for (int j = 0; j < 4; ++j) {
    float bb = b1[j * 16 + n0];
#pragma unroll
    for (int r = 0; r < 8; ++r)
      sHid[wave][(r + 8 * half) * 64 + j * 16 + n0] = frelu(a1[j][r] + bb);
  }
  __syncthreads();

  v8f a2[4] = {};
  gemm_tile<64>(sHid[wave], W2, a2);
#pragma unroll
  for (int j = 0; j < 4; ++j) {
    float bb = b2[j * 16 + n0];
#pragma unroll
    for (int r = 0; r < 8; ++r)
      sEmb[wave][(r + 8 * half) * 64 + j * 16 + n0] = a2[j][r] + bb;
  }
  __syncthreads();

  v8f a3[4] = {};
  gemm_tile<64>(sEmb[wave], Wg, a3);
#pragma unroll
  for (int j = 0; j < 4; ++j)
#pragma unroll
    for (int r = 0; r < 8; ++r) {
      int row = base + r + 8 * half;
      if (row < n) hgcn[(long long)row * 64 + j * 16 + n0] = a3[j][r];
    }
}

// ----------------------------------------------------------- GCN edge scatter
__global__ void k_gcn_scatter(const int* __restrict__ ei,
                              const float* __restrict__ hgcn,
                              const float* __restrict__ dinv,
                              float* __restrict__ accum, int e) {
  int tid = blockIdx.x * blockDim.x + threadIdx.x;
  if (tid >= e) return;
  int s = ei[tid], d = ei[e + tid];
  float nm = dinv[s] * dinv[d];
  const float4* hs = (const float4*)(hgcn + (long long)s * 64);
  float* ad = accum + (long long)d * 64;
#pragma unroll
  for (int q = 0; q < 16; ++q) {
    float4 v = hs[q];
    atomicAdd(ad + q * 4 + 0, v.x * nm);
    atomicAdd(ad + q * 4 + 1, v.y * nm);
    atomicAdd(ad + q * 4 + 2, v.z * nm);
    atomicAdd(ad + q * 4 + 3, v.w * nm);
  }
}

// ------------------------- GCN finish (+self loop,+bias,relu) -> gat_W (WMMA)
__global__ void __launch_bounds__(128) k_gcn_finish(
    const float* __restrict__ hgcn, float* __restrict__ accum,
    const float* __restrict__ dinv, const float* __restrict__ gcn_b,
    const float* __restrict__ gatW, const float* __restrict__ attS,
    const float* __restrict__ attD, float* __restrict__ hgat,
    float* __restrict__ a_s, float* __restrict__ a_d,
    unsigned* __restrict__ mmax, float* __restrict__ zsum, int n) {
  __shared__ float sX[4][16 * 64];
  __shared__ float sH[4][16 * 64];
  const int wave = threadIdx.x >> 5;
  const int lane = threadIdx.x & 31;
  const int half = lane >> 4;
  const int n0 = lane & 15;
  const int base = (blockIdx.x * 4 + wave) * 16;

  for (int idx = lane; idx < 1024; idx += 32) {
    int m = idx >> 6, c = idx & 63;
    int node = base + m;
    bool valid = node < n;
    if (!valid) node = n - 1;
    float di = dinv[node];
    long long off = (long long)node * 64 + c;
    float v = accum[off] + hgcn[off] * di * di + gcn_b[c];
    sX[wave][idx] = frelu(v);
    if (valid) accum[off] = 0.f;  // recycle buffer for GAT aggregation
  }
  __syncthreads();

  v8f acc[4] = {};
  gemm_tile<64>(sX[wave], gatW, acc);
#pragma unroll
  for (int j = 0; j < 4; ++j)
#pragma unroll
    for (int r = 0; r < 8; ++r) {
      float v = acc[j][r];
      int rr = r + 8 * half;
      sH[wave][rr * 64 + j * 16 + n0] = v;
      int row = base + rr;
      if (row < n) hgat[(long long)row * 64 + j * 16 + n0] = v;
    }
  __syncthreads();

  if (lane < 16) {
    int node = base + lane;
    if (node < n) {
      float ds = 0.f, dd = 0.f;
      for (int k = 0; k < 64; ++k) {
        float v = sH[wave][lane * 64 + k];
        ds += v * attS[k];
        dd += v * attD[k];
      }
      a_s[node] = ds;
      a_d[node] = dd;
      mmax[node] = fenc(leaky(ds + dd));  // init with self-loop logit
      zsum[node] = 0.f;
    }
  }
}

// ----------------------------------------------------------- GAT softmax passes
__global__ void k_gat_max(const int* __restrict__ ei, const float* __restrict__ a_s,
                          const float* __restrict__ a_d, unsigned* __restrict__ mmax,
                          int e) {
  int tid = blockIdx.x * blockDim.x + threadIdx.x;
  if (tid >= e) return;
  int s = ei[tid], d = ei[e + tid];
  atomicMax(&mmax[d], fenc(leaky(a_s[s] + a_d[d])));
}

__global__ void k_gat_exp(const int* __restrict__ ei, const float* __restrict__ a_s,
                          const float* __restrict__ a_d,
                          const unsigned* __restrict__ mmax,
                          float* __restrict__ zsum, float* __restrict__ exbuf, int e) {
  int tid = blockIdx.x * blockDim.x + threadIdx.x;
  if (tid >= e) return;
  int s = ei[tid], d = ei[e + tid];
  float ex = __expf(leaky(a_s[s] + a_d[d]) - fdec(mmax[d]));
  exbuf[tid] = ex;
  atomicAdd(&zsum[d], ex);
}

__global__ void k_z_self(const float* __restrict__ a_s, const float* __restrict__ a_d,
                         const unsigned* __restrict__ mmax, float* __restrict__ zsum,
                         int n) {
  int tid = blockIdx.x * blockDim.x + threadIdx.x;
  if (tid >= n) return;
  zsum[tid] += __expf(leaky(a_s[tid] + a_d[tid]) - fdec(mmax[tid]));
}

__global__ void k_gat_scatter(const int* __restrict__ ei,
                              const float* __restrict__ hgat,
                              const float* __restrict__ exbuf,
                              const float* __restrict__ zsum,
                              float* __restrict__ accum, int e) {
  int tid = blockIdx.x * blockDim.x + threadIdx.x;
  if (tid >= e) return;
  int s = ei[tid], d = ei[e + tid];
  float alpha = exbuf[tid] / zsum[d];
  const float4* hs = (const float4*)(hgat + (long long)s * 64);
  float* ad = accum + (long long)d * 64;
#pragma unroll
  for (int q = 0; q < 16; ++q) {
    float4 v = hs[q];
    atomicAdd(ad + q * 4 + 0, v.x * alpha);
    atomicAdd(ad + q * 4 + 1, v.y * alpha);
    atomicAdd(ad + q * 4 + 2, v.z * alpha);
    atomicAdd(ad + q * 4 + 3, v.w * alpha);
  }
}

// ------------------------------------- fused head: gate, residual, speed (WMMA)
__global__ void __launch_bounds__(128) k_head(
    const float* __restrict__ H, const float* __restrict__ accum,
    const float* __restrict__ hgat, const float* __restrict__ a_s,
    const float* __restrict__ a_d, const unsigned* __restrict__ mmax,
    const float* __restrict__ zsum, const float* __restrict__ gat_b,
    const float* __restrict__ gateW, const float* __restrict__ gate_b,
    const float* __restrict__ resW1, const float* __restrict__ res_b1,
    const float* __restrict__ resW2, const float* __restrict__ res_b2,
    const float* __restrict__ spW1, const float* __restrict__ sp_b1,
    const float* __restrict__ spW2, const float* __restrict__ sp_b2,
    float* __restrict__ out_delta, float* __restrict__ out_H,
    float* __restrict__ out_pred, int n) {
  __shared__ float sF[4][16 * 128];  // fusion tile [H | diffused]
  __shared__ float sT[4][16 * 64];   // hidden, then H_final
  __shared__ float sA[4][16];        // self-loop alpha
  const int wave = threadIdx.x >> 5;
  const int lane = threadIdx.x & 31;
  const int half = lane >> 4;
  const int n0 = lane & 15;
  const int base = (blockIdx.x * 4 + wave) * 16;

  if (lane < 16) {
    int node = base + lane;
    if (node >= n) node = n - 1;
    float es = leaky(a_s[node] + a_d[node]);
    sA[wave][lane] = __expf(es - fdec(mmax[node])) / zsum[node];
  }
  __syncthreads();

  for (int idx = lane; idx < 2048; idx += 32) {
    int m = idx >> 7, c = idx & 127;
    int node = base + m;
    if (node >= n) node = n - 1;
    float v;
    if (c < 64) {
      v = H[(long long)node * 64 + c];
    } else {
      int cc = c - 64;
      long long off = (long long)node * 64 + cc;
      v = frelu(accum[off] + hgat[off] * sA[wave][m] + gat_b[cc]);
    }
    sF[wave][idx] = v;
  }
  __syncthreads();

  v8f gacc[4] = {};
  gemm_tile<128>(sF[wave], gateW, gacc);
  v8f hacc[4] = {};
  gemm_tile<128>(sF[wave], resW1, hacc);
#pragma unroll
  for (int j = 0; j < 4; ++j) {
    float bg = gate_b[j * 16 + n0];
    float bh = res_b1[j * 16 + n0];
#pragma unroll
    for (int r = 0; r < 8; ++r) {
      gacc[j][r] = 1.f / (1.f + __expf(-(gacc[j][r] + bg)));
      sT[wave][(r + 8 * half) * 64 + j * 16 + n0] = frelu(hacc[j][r] + bh);
    }
  }
  __syncthreads();

  v8f dacc[4] = {};
  gemm_tile<64>(sT[wave], resW2, dacc);
  __syncthreads();  // all lanes done reading sT before we overwrite it

#pragma unroll
  for (int j = 0; j < 4; ++j) {
    float bd = res_b2[j * 16 + n0];
#pragma unroll
    for (int r = 0; r < 8; ++r) {
      float delta = gacc[j][r] * (dacc[j][r] + bd);
      int rr = r + 8 * half;
      float hfin = sF[wave][rr * 128 + j * 16 + n0] + delta;
      sT[wave][rr * 64 + j * 16 + n0] = hfin;
      int row = base + rr;
      if (row < n) {
        out_delta[(long long)row * 64 + j * 16 + n0] = delta;
        out_H[(long long)row * 64 + j * 16 + n0] = hfin;
      }
    }
  }
  __syncthreads();

  if (lane < 16) {
    int node = base + lane;
    if (node < n) {
      float p = sp_b2[0];
      for (int q = 0; q < 32; ++q) {
        float s = sp_b1[q];
        for (int k = 0; k < 64; ++k) s += sT[wave][lane * 64 + k] * spW1[k * 32 + q];
        p += frelu(s) * spW2[q];
      }
      out_pred[node] = p;
    }
  }
}

// -------------------------------------------------------------------- launcher
extern "C" void kernel_launch(void* const* d_in, const int* in_sizes, int n_in,
                              void* d_out, int out_size, void* d_ws, size_t ws_size,
                              hipStream_t stream) {
  const float* H = (const float*)d_in[0];
  const float* ev = (const float*)d_in[1];
  const int* ei = (const int*)d_in[2];
  const float* encW1 = (const float*)d_in[3];
  const float* encb1 = (const float*)d_in[4];
  const float* encW2 = (const float*)d_in[5];
  const float* encb2 = (const float*)d_in[6];
  const float* gcnW = (const float*)d_in[7];
  const float* gcnb = (const float*)d_in[8];
  const float* gatW = (const float*)d_in[9];
  const float* attS = (const float*)d_in[10];
  const float* attD = (const float*)d_in[11];
  const float* gatb = (const float*)d_in[12];
  const float* gateW = (const float*)d_in[13];
  const float* gateb = (const float*)d_in[14];
  const float* resW1 = (const float*)d_in[15];
  const float* resb1 = (const float*)d_in[16];
  const float* resW2 = (const float*)d_in[17];
  const float* resb2 = (const float*)d_in[18];
  const float* spW1 = (const float*)d_in[19];
  const float* spb1 = (const float*)d_in[20];
  const float* spW2 = (const float*)d_in[21];
  const float* spb2 = (const float*)d_in[22];

  const int n = in_sizes[0] / 64;
  const int e = in_sizes[2] / 2;

  float* ws = (float*)d_ws;
  float* hgcn = ws;   ws += (size_t)n * 64;
  float* accum = ws;  ws += (size_t)n * 64;
  float* hgat = ws;   ws += (size_t)n * 64;
  float* dinv = ws;   ws += n;  // degrees, then D^{-1/2} in place
  float* a_s = ws;    ws += n;
  float* a_d = ws;    ws += n;
  unsigned* mmax = (unsigned*)ws; ws += n;
  float* zsum = ws;   ws += n;
  float* exbuf = ws;  ws += e;

  float* out_delta = (float*)d_out;
  float* out_H = out_delta + (size_t)n * 64;
  float* out_pred = out_H + (size_t)n * 64;

  const int tiles = (n + 15) / 16;
  const int wblocks = (tiles + 3) / 4;  // 4 waves/block, 1 tile/wave
  const int eblocks = (e + 255) / 256;
  const int nblocks = (n + 255) / 256;
  const long long initT = (long long)n * 64;
  const int iblocks = (int)((initT + 255) / 256);

  k_init<<<iblocks, 256, 0, stream>>>(accum, dinv, n);
  k_deg<<<eblocks, 256, 0, stream>>>(ei, dinv, e);
  k_dinv<<<nblocks, 256, 0, stream>>>(dinv, n);
  k_encoder<<<wblocks, 128, 0, stream>>>(ev, encW1, encb1, encW2, encb2, gcnW,
                                         hgcn, n);
  k_gcn_scatter<<<eblocks, 256, 0, stream>>>(ei, hgcn, dinv, accum, e);
  k_gcn_finish<<<wblocks, 128, 0, stream>>>(hgcn, accum, dinv, gcnb, gatW, attS,
                                            attD, hgat, a_s, a_d, mmax, zsum, n);
  k_gat_max<<<eblocks, 256, 0, stream>>>(ei, a_s, a_d, mmax, e);
  k_gat_exp<<<eblocks, 256, 0, stream>>>(ei, a_s, a_d, mmax, zsum, exbuf, e);
  k_z_self<<<nblocks, 256, 0, stream>>>(a_s, a_d, mmax, zsum, n);
  k_gat_scatter<<<eblocks, 256, 0, stream>>>(ei, hgat, exbuf, zsum, accum, e);
  k_head<<<wblocks, 128, 0, stream>>>(H, accum, hgat, a_s, a_d, mmax, zsum, gatb,
                                      gateW, gateb, resW1, resb1, resW2, resb2,
                                      spW1, spb1, spW2, spb2, out_delta, out_H,
                                      out_pred, n);
}